// SoftHd_76493367541802
// MI455X (gfx1250) — compile-verified
//
#include <hip/hip_runtime.h>

#define N_MAX   1536
#define BZ      64
#define DF      64
#define BIG_F   1e30f
#define NWAVES  8
#define THREADS (NWAVES * 32)
#define SPLIT   6

typedef float v2f __attribute__((ext_vector_type(2)));
typedef float v8f __attribute__((ext_vector_type(8)));

// Order-preserving map float -> uint so we can use integer atomic min.
__device__ __forceinline__ unsigned fkey(float f) {
    unsigned u = __float_as_uint(f);
    return (u & 0x80000000u) ? ~u : (u | 0x80000000u);
}
__device__ __forceinline__ float finv(unsigned k) {
    unsigned u = (k & 0x80000000u) ? (k ^ 0x80000000u) : ~k;
    return __uint_as_float(u);
}

__global__ void softhd_init(unsigned* __restrict__ colmin, float* __restrict__ rowsum) {
    int i = blockIdx.x * blockDim.x + threadIdx.x;
    if (i < BZ * N_MAX) colmin[i] = fkey(BIG_F);
    if (i < BZ) rowsum[i] = 0.0f;
}

__global__ __launch_bounds__(THREADS)
void softhd_main(const float* __restrict__ x1, const float* __restrict__ x2,
                 const int* __restrict__ sz1, const int* __restrict__ sz2,
                 unsigned* __restrict__ g_colmin, float* __restrict__ g_rowsum) {
    __shared__ float    s_n1[N_MAX];
    __shared__ float    s_n2[N_MAX];
    __shared__ unsigned s_colmin[N_MAX];
    __shared__ float    s_rowsum;

    const int b     = blockIdx.x;
    const int split = blockIdx.y;
    const int tid   = threadIdx.x;
    const int N = sz1[b];
    const int M = sz2[b];

    // Exclusive prefix offsets into the ragged flat arrays (uniform scalar loop).
    int off1 = 0, off2 = 0;
    for (int i = 0; i < b; ++i) { off1 += sz1[i]; off2 += sz2[i]; }
    const float* X1 = x1 + (size_t)off1 * DF;
    const float* X2 = x2 + (size_t)off2 * DF;

    if (tid == 0) s_rowsum = 0.0f;
    for (int i = tid; i < N_MAX; i += THREADS) s_colmin[i] = fkey(BIG_F);

    // Per-row squared norms into LDS.
    for (int r = tid; r < N; r += THREADS) {
        const float4* p = (const float4*)(X1 + (size_t)r * DF);
        float s = 0.0f;
#pragma unroll
        for (int q = 0; q < DF / 4; ++q) {
            float4 v = p[q];
            s += v.x * v.x + v.y * v.y + v.z * v.z + v.w * v.w;
        }
        s_n1[r] = s;
    }
    for (int r = tid; r < M; r += THREADS) {
        const float4* p = (const float4*)(X2 + (size_t)r * DF);
        float s = 0.0f;
#pragma unroll
        for (int q = 0; q < DF / 4; ++q) {
            float4 v = p[q];
            s += v.x * v.x + v.y * v.y + v.z * v.z + v.w * v.w;
        }
        s_n2[r] = s;
    }
    __syncthreads();

    const int wave = tid >> 5;
    const int lane = tid & 31;
    const int lo   = lane & 15;   // A row / B col / D col within tile
    const int hi   = lane >> 4;   // half selector
    const int k0   = hi * 2;      // K sub-offset for A/B fragments
    const int mtiles = (M + 15) >> 4;

    for (int strip = split * NWAVES + wave; strip * 16 < N; strip += SPLIT * NWAVES) {
        const int n0 = strip * 16;

        // A fragments for this 16-row strip: fixed across the m-loop.
        int arow = n0 + lo; if (arow >= N) arow = N - 1;
        const float* ap = X1 + (size_t)arow * DF + k0;
        v2f afrag[16];
#pragma unroll
        for (int f = 0; f < 16; ++f) afrag[f] = *(const v2f*)(ap + 4 * f);

        // Norms + validity for this lane's 8 D-tile rows (M = v + 8*hi).
        float n1v[8];
        bool  rvalid[8];
#pragma unroll
        for (int v = 0; v < 8; ++v) {
            int r = n0 + hi * 8 + v;
            rvalid[v] = (r < N);
            if (r >= N) r = N - 1;
            n1v[v] = s_n1[r];
        }

        float rmin[8];
#pragma unroll
        for (int v = 0; v < 8; ++v) rmin[v] = BIG_F;

        for (int mt = 0; mt < mtiles; ++mt) {
            const int m0   = mt * 16;
            const int mcol = m0 + lo;
            const bool mval = (mcol < M);
            const int  mc   = mval ? mcol : (M - 1);

            const float* bp = X2 + (size_t)mc * DF + k0;
            v2f bfrag[16];
#pragma unroll
            for (int f = 0; f < 16; ++f) bfrag[f] = *(const v2f*)(bp + 4 * f);

            v8f c = {};
#pragma unroll
            for (int f = 0; f < 16; ++f)
                c = __builtin_amdgcn_wmma_f32_16x16x4_f32(
                        false, afrag[f], false, bfrag[f], (short)0, c, false, false);

            const float n2v = s_n2[mc];
            float cmin = BIG_F;
#pragma unroll
            for (int v = 0; v < 8; ++v) {
                float val = n1v[v] + n2v - 2.0f * c[v];
                if (rvalid[v]) cmin = fminf(cmin, val);
                if (mval)      rmin[v] = fminf(rmin[v], val);
            }
            if (mval) atomicMin(&s_colmin[mcol], fkey(cmin));
        }

        // Cross-lane min over the 16 columns (stays within each 16-lane half).
#pragma unroll
        for (int v = 0; v < 8; ++v) {
            float r = rmin[v];
            r = fminf(r, __shfl_xor(r, 1, 32));
            r = fminf(r, __shfl_xor(r, 2, 32));
            r = fminf(r, __shfl_xor(r, 4, 32));
            r = fminf(r, __shfl_xor(r, 8, 32));
            rmin[v] = r;
        }
        if (lo == 0) {
            float s = 0.0f;
#pragma unroll
            for (int v = 0; v < 8; ++v)
                if (rvalid[v]) s += rmin[v];
            atomicAdd(&s_rowsum, s);
        }
    }
    __syncthreads();

    // Merge this workgroup's partial results into per-batch globals.
    for (int m = tid; m < M; m += THREADS)
        atomicMin(&g_colmin[b * N_MAX + m], s_colmin[m]);
    if (tid == 0) atomicAdd(&g_rowsum[b], s_rowsum);
}

__global__ __launch_bounds__(256)
void softhd_final(const unsigned* __restrict__ colmin, const float* __restrict__ rowsum,
                  const int* __restrict__ sz1, const int* __restrict__ sz2,
                  float* __restrict__ out) {
    __shared__ float red[256];
    const int b = blockIdx.x, tid = threadIdx.x;
    const int N = sz1[b], M = sz2[b];
    float s = 0.0f;
    for (int m = tid; m < M; m += 256) s += finv(colmin[b * N_MAX + m]);
    red[tid] = s;
    __syncthreads();
    for (int w = 128; w > 0; w >>= 1) {
        if (tid < w) red[tid] += red[tid + w];
        __syncthreads();
    }
    if (tid == 0) out[b] = (red[0] + rowsum[b]) / (float)(N + M);
}

extern "C" void kernel_launch(void* const* d_in, const int* in_sizes, int n_in,
                              void* d_out, int out_size, void* d_ws, size_t ws_size,
                              hipStream_t stream) {
    (void)in_sizes; (void)n_in; (void)out_size; (void)ws_size;
    const float* x1  = (const float*)d_in[0];
    const float* x2  = (const float*)d_in[1];
    const int*   sz1 = (const int*)d_in[2];
    const int*   sz2 = (const int*)d_in[3];

    unsigned* colmin = (unsigned*)d_ws;
    float*    rowsum = (float*)((char*)d_ws + (size_t)BZ * N_MAX * sizeof(unsigned));
    float*    out    = (float*)d_out;

    softhd_init<<<(BZ * N_MAX + 255) / 256, 256, 0, stream>>>(colmin, rowsum);
    softhd_main<<<dim3(BZ, SPLIT), THREADS, 0, stream>>>(x1, x2, sz1, sz2, colmin, rowsum);
    softhd_final<<<BZ, 256, 0, stream>>>(colmin, rowsum, sz1, sz2, out);
}